// _ImplicitQSRNetwork_3461743640831
// MI455X (gfx1250) — compile-verified
//
#include <hip/hip_runtime.h>
#include <cmath>

// ---------------------------------------------------------------------------
// ImplicitQSRNetwork forward for MI455X (gfx1250, wave32).
// Large GEMMs: V_WMMA_F32_16X16X4_F32 with LDS-staged, double-buffered tiles
// fed by GLOBAL_LOAD_ASYNC_TO_LDS_B128 (ASYNCcnt path) when available.
// ---------------------------------------------------------------------------

constexpr int kNX = 512;
constexpr int kNY = 64;
constexpr int kNU = 64;
constexpr int kNQ = 256;
constexpr int kNB = 8192;
constexpr int kSD = 64;          // SDIM
constexpr int kN  = 1280;        // 2*NX + NQ
constexpr float kEPS = 1e-3f;

typedef __attribute__((ext_vector_type(2))) float v2f;
typedef __attribute__((ext_vector_type(8))) float v8f;

#if defined(__AMDGCN__) && __has_builtin(__builtin_amdgcn_wmma_f32_16x16x4_f32)
#define HAVE_WMMA_F32 1
#else
#define HAVE_WMMA_F32 0
#endif

#if defined(__AMDGCN__) && __has_builtin(__builtin_amdgcn_global_load_async_to_lds_b128) && \
    __has_builtin(__builtin_amdgcn_s_wait_asynccnt)
#define HAVE_ASYNC_LDS 1
#else
#define HAVE_ASYNC_LDS 0
#endif

#if HAVE_ASYNC_LDS
// The builtin's params (from the clang diagnostic) are pointers to a GCC-style
// int4 vector: 'int __attribute__((vector_size(16)))' in AS1 (printed
// '__device__') for the global side, AS3 for the LDS side.
typedef int b128v __attribute__((vector_size(16)));
typedef __attribute__((address_space(1))) b128v* g_b128p;
typedef __attribute__((address_space(3))) b128v* l_b128p;
#endif

// Copy 16 bytes global -> LDS. Async (ASYNCcnt-tracked) when the gfx1250
// builtin exists; otherwise a synchronous float4 load + ds_store_b128.
__device__ __forceinline__ void copy16_g2l(const float* __restrict__ g, float* l)
{
#if HAVE_ASYNC_LDS
  __builtin_amdgcn_global_load_async_to_lds_b128(
      (g_b128p)(unsigned long long)g,
      (l_b128p)(unsigned int)(unsigned long long)l,
      0, 0);
#else
  *(float4*)l = *(const float4*)g;
#endif
}

__device__ __forceinline__ void wait_async_all()
{
#if HAVE_ASYNC_LDS
  __builtin_amdgcn_s_wait_asynccnt(0);
#endif
}

// ---------------------------------------------------------------------------
// WMMA GEMM-NT:  C[M,N] = alpha * A[M,K] * B[N,K]^T + beta*C + deps*I
// Block: 256 threads = 8 waves; block tile 128(M) x 64(N); K staged in
// double-buffered LDS chunks of 32. Each wave owns a 16x64 strip
// (4 accumulators; A-fragment reused across the 4 N-subtiles).
// Requires: M % 128 == 0, N % 64 == 0, K % 32 == 0 (true for all call sites).
// LDS row stride 36 floats: rows 16B-aligned for B128, and 36*r mod 64 is
// injective for r=0..15 -> conflict-free ds_load_b64 fragment reads.
// ---------------------------------------------------------------------------
constexpr int TM  = 128;
constexpr int TN  = 64;
constexpr int TK  = 32;
constexpr int LSR = 36;   // padded LDS row stride in floats

__global__ __launch_bounds__(256) void wmma_gemm_nt(
    const float* __restrict__ A, const float* __restrict__ B,
    float* __restrict__ C, int M, int N, int K,
    int lda, int ldb, int ldc, float alpha, float beta, float deps)
{
  const int tid  = threadIdx.x;
  const int lane = tid & 31;
  const int wv   = tid >> 5;
  const int hl   = lane >> 4;
  const int lh   = lane & 15;
  const int mblk = blockIdx.y * TM;
  const int nblk = blockIdx.x * TN;

#if HAVE_WMMA_F32
  __shared__ __align__(16) float sA[2][TM * LSR];
  __shared__ __align__(16) float sB[2][TN * LSR];

  auto stage = [&](int k0, int sel) {
    // A tile: 128x32 floats = 1024 float4, 4 per thread
#pragma unroll
    for (int it = 0; it < 4; ++it) {
      const int e = tid + 256 * it;
      const int row = e >> 3, c4 = e & 7;
      copy16_g2l(A + (size_t)(mblk + row) * lda + k0 + 4 * c4,
                 &sA[sel][row * LSR + 4 * c4]);
    }
    // B tile: 64x32 floats = 512 float4, 2 per thread
#pragma unroll
    for (int it = 0; it < 2; ++it) {
      const int e = tid + 256 * it;
      const int row = e >> 3, c4 = e & 7;
      copy16_g2l(B + (size_t)(nblk + row) * ldb + k0 + 4 * c4,
                 &sB[sel][row * LSR + 4 * c4]);
    }
  };

  v8f acc[4];
#pragma unroll
  for (int j = 0; j < 4; ++j)
#pragma unroll
    for (int r = 0; r < 8; ++r) acc[j][r] = 0.0f;

  stage(0, 0);
  wait_async_all();
  __syncthreads();

  const int nch = K / TK;
  for (int kc = 0; kc < nch; ++kc) {
    const int cur = kc & 1;
    if (kc + 1 < nch) stage((kc + 1) * TK, cur ^ 1);

    const float* la  = &sA[cur][(wv * 16 + lh) * LSR];
    const float* lb0 = &sB[cur][( 0 + lh) * LSR];
    const float* lb1 = &sB[cur][(16 + lh) * LSR];
    const float* lb2 = &sB[cur][(32 + lh) * LSR];
    const float* lb3 = &sB[cur][(48 + lh) * LSR];

#pragma unroll
    for (int k = 0; k < TK; k += 4) {
      const int kk = k + 2 * hl;
      v2f a;  a.x  = la[kk];  a.y  = la[kk + 1];
      v2f b0; b0.x = lb0[kk]; b0.y = lb0[kk + 1];
      v2f b1; b1.x = lb1[kk]; b1.y = lb1[kk + 1];
      v2f b2; b2.x = lb2[kk]; b2.y = lb2[kk + 1];
      v2f b3; b3.x = lb3[kk]; b3.y = lb3[kk + 1];
      acc[0] = __builtin_amdgcn_wmma_f32_16x16x4_f32(false, a, false, b0, (short)0, acc[0], false, false);
      acc[1] = __builtin_amdgcn_wmma_f32_16x16x4_f32(false, a, false, b1, (short)0, acc[1], false, false);
      acc[2] = __builtin_amdgcn_wmma_f32_16x16x4_f32(false, a, false, b2, (short)0, acc[2], false, false);
      acc[3] = __builtin_amdgcn_wmma_f32_16x16x4_f32(false, a, false, b3, (short)0, acc[3], false, false);
    }
    wait_async_all();
    __syncthreads();
  }

#pragma unroll
  for (int j = 0; j < 4; ++j) {
    const int col = nblk + 16 * j + lh;
#pragma unroll
    for (int r = 0; r < 8; ++r) {
      const int row = mblk + wv * 16 + r + 8 * hl;
      const size_t off = (size_t)row * ldc + col;
      float v = alpha * acc[j][r];
      if (beta != 0.0f) v += beta * C[off];
      if (deps != 0.0f && row == col) v += deps;
      C[off] = v;
    }
  }
#else
  // Scalar fallback (also used by the host-side parse); same output mapping.
  for (int j = 0; j < 4; ++j) {
    const int col = nblk + 16 * j + lh;
    for (int r = 0; r < 8; ++r) {
      const int row = mblk + wv * 16 + r + 8 * hl;
      float s = 0.0f;
      for (int k = 0; k < K; ++k)
        s += A[(size_t)row * lda + k] * B[(size_t)col * ldb + k];
      const size_t off = (size_t)row * ldc + col;
      float v = alpha * s;
      if (beta != 0.0f) v += beta * C[off];
      if (deps != 0.0f && row == col) v += deps;
      C[off] = v;
    }
  }
#endif
}

// ---------------------------------------------------------------------------
// Scalar small GEMM for 64-sized frame algebra (negligible FLOPs):
// C[i,j] = alpha * sum_k opA(i,k)*opB(k,j) + beta * opC(i,j)
// ---------------------------------------------------------------------------
__global__ void gemm_small(const float* __restrict__ A, const float* __restrict__ B,
                           const float* __restrict__ Cin, float* __restrict__ Cout,
                           int M, int N, int K, int lda, int ldb, int ldci, int ldc,
                           float alpha, float beta, int tA, int tB, int tC)
{
  const int idx = blockIdx.x * blockDim.x + threadIdx.x;
  if (idx >= M * N) return;
  const int i = idx / N, j = idx % N;
  float s = 0.0f;
  for (int k = 0; k < K; ++k) {
    const float av = tA ? A[(size_t)k * lda + i] : A[(size_t)i * lda + k];
    const float bv = tB ? B[(size_t)j * ldb + k] : B[(size_t)k * ldb + j];
    s += av * bv;
  }
  float c = 0.0f;
  if (beta != 0.0f)
    c = tC ? Cin[(size_t)j * ldci + i] : Cin[(size_t)i * ldci + j];
  Cout[(size_t)i * ldc + j] = alpha * s + beta * c;
}

// M = X3 X3^T + Y3 - Y3^T + eps I ;  IpM = I+M ; ImM = I-M
__global__ void k_build_M(const float* __restrict__ X3, const float* __restrict__ Y3,
                          float* __restrict__ IpM, float* __restrict__ ImM)
{
  const int idx = blockIdx.x * blockDim.x + threadIdx.x;
  if (idx >= kSD * kSD) return;
  const int i = idx / kSD, j = idx % kSD;
  float s = 0.0f;
  for (int k = 0; k < kSD; ++k) s += X3[i * kSD + k] * X3[j * kSD + k];
  const float m  = s + Y3[i * kSD + j] - Y3[j * kSD + i] + ((i == j) ? kEPS : 0.0f);
  const float id = (i == j) ? 1.0f : 0.0f;
  IpM[idx] = id + m;
  ImM[idx] = id - m;
}

// 64x64 Gauss-Jordan inverse in LDS (1 block, 256 threads)
__global__ __launch_bounds__(256) void k_gj64(const float* __restrict__ A, int lda,
                                              float* __restrict__ Ainv)
{
  __shared__ float aug[kSD][2 * kSD + 2];
  __shared__ float fcol[kSD];
  const int tid = threadIdx.x;
  for (int idx = tid; idx < kSD * 2 * kSD; idx += 256) {
    const int i = idx >> 7, j = idx & 127;
    aug[i][j] = (j < kSD) ? A[(size_t)i * lda + j] : ((j - kSD) == i ? 1.0f : 0.0f);
  }
  __syncthreads();
  for (int kp = 0; kp < kSD; ++kp) {
    const float pin = 1.0f / aug[kp][kp];
    if (tid < kSD) fcol[tid] = (tid == kp) ? 0.0f : aug[tid][kp];
    __syncthreads();
    if (tid < 2 * kSD) aug[kp][tid] *= pin;
    __syncthreads();
    for (int idx = tid; idx < kSD * 2 * kSD; idx += 256) {
      const int i = idx >> 7, j = idx & 127;
      aug[i][j] -= fcol[i] * aug[kp][j];
    }
    __syncthreads();
  }
  for (int idx = tid; idx < kSD * kSD; idx += 256) {
    const int i = idx >> 6, j = idx & 63;
    Ainv[idx] = aug[i][kSD + j];
  }
}

// 64x64 Cholesky (lower) in LDS (1 block, 64 threads)
__global__ __launch_bounds__(64) void k_chol64(const float* __restrict__ A,
                                               float* __restrict__ L)
{
  __shared__ float a[kSD][kSD + 1];
  const int tid = threadIdx.x;
  for (int i = 0; i < kSD; ++i) a[i][tid] = A[i * kSD + tid];
  __syncthreads();
  for (int k = 0; k < kSD; ++k) {
    if (tid == k) a[k][k] = sqrtf(a[k][k]);
    __syncthreads();
    if (tid > k) a[tid][k] /= a[k][k];
    __syncthreads();
    if (tid > k)
      for (int i = tid; i < kSD; ++i) a[i][tid] -= a[i][k] * a[tid][k];
    __syncthreads();
  }
  for (int i = 0; i < kSD; ++i) L[i * kSD + tid] = (tid <= i) ? a[i][tid] : 0.0f;
}

// Forward substitution: X = L^{-1} B (64x64), one thread per RHS column.
__global__ __launch_bounds__(64) void k_trsm64(const float* __restrict__ L,
                                               const float* __restrict__ B,
                                               float* __restrict__ X)
{
  __shared__ float Ls[kSD][kSD + 1];
  __shared__ float Xs[kSD][kSD + 1];
  const int c = threadIdx.x;
  for (int i = 0; i < kSD; ++i) Ls[i][c] = L[i * kSD + c];
  __syncthreads();
  for (int i = 0; i < kSD; ++i) {
    float s = B[i * kSD + c];
    for (int j = 0; j < i; ++j) s -= Ls[i][j] * Xs[j][c];
    Xs[i][c] = s / Ls[i][i];
  }
  for (int i = 0; i < kSD; ++i) X[i * kSD + c] = Xs[i][c];
}

// Rc += T + T^T
__global__ void k_add_sym(float* __restrict__ Rc, const float* __restrict__ T)
{
  const int idx = blockIdx.x * blockDim.x + threadIdx.x;
  if (idx >= kSD * kSD) return;
  const int i = idx / kSD, j = idx % kSD;
  Rc[idx] += T[idx] + T[j * kSD + i];
}

// V = [C2_t^T ; D21_t^T ; B2] (1280x64), vecC = [C2^T ; D21^T ; 0]
__global__ void k_pack_VC(const float* __restrict__ C2t, const float* __restrict__ D21t,
                          const float* __restrict__ B2, const float* __restrict__ C2,
                          const float* __restrict__ D21,
                          float* __restrict__ V, float* __restrict__ vecC)
{
  const int idx = blockIdx.x * blockDim.x + threadIdx.x;
  if (idx >= kN * kSD) return;
  const int i = idx / kSD, j = idx % kSD;
  float v, c;
  if (i < kNX)           { v = C2t[(size_t)j * kNX + i];          c = C2[(size_t)j * kNX + i]; }
  else if (i < kNX + kNQ){ const int r = i - kNX;
                           v = D21t[(size_t)j * kNQ + r];         c = D21[(size_t)j * kNQ + r]; }
  else                   { const int r = i - kNX - kNQ;
                           v = B2[(size_t)r * kSD + j];           c = 0.0f; }
  V[idx] = v; vecC[idx] = c;
}

// E = 0.5*(H11 + (1/alpha^2) P + Y1 - Y1^T), 1/alpha^2 = 4
__global__ void k_build_E(const float* __restrict__ H, const float* __restrict__ Y1,
                          float* __restrict__ E)
{
  const int idx = blockIdx.x * blockDim.x + threadIdx.x;
  if (idx >= kNX * kNX) return;
  const int i = idx / kNX, j = idx % kNX;
  const float p = H[(size_t)(768 + i) * kN + 768 + j];
  E[idx] = 0.5f * (H[(size_t)i * kN + j] + 4.0f * p + Y1[idx] - Y1[(size_t)j * kNX + i]);
}

__global__ void k_lam(const float* __restrict__ H, float* __restrict__ lam)
{
  const int k = blockIdx.x * blockDim.x + threadIdx.x;
  if (k >= kNQ) return;
  lam[k] = 0.5f * H[(size_t)(kNX + k) * kN + kNX + k];
}

__global__ void k_build_D11(const float* __restrict__ H, const float* __restrict__ lam,
                            float* __restrict__ D11)
{
  const int idx = blockIdx.x * blockDim.x + threadIdx.x;
  if (idx >= kNQ * kNQ) return;
  const int k = idx / kNQ, j = idx % kNQ;
  D11[idx] = (j < k) ? -H[(size_t)(kNX + k) * kN + kNX + j] / lam[k] : 0.0f;
}

__global__ void k_build_C1(const float* __restrict__ H, float* __restrict__ C1)
{
  const int idx = blockIdx.x * blockDim.x + threadIdx.x;
  if (idx >= kNQ * kNX) return;
  const int k = idx / kNX, j = idx % kNX;
  C1[idx] = -H[(size_t)(kNX + k) * kN + j];
}

__global__ void k_build_D12(const float* __restrict__ D12t, const float* __restrict__ lam,
                            float* __restrict__ D12)
{
  const int idx = blockIdx.x * blockDim.x + threadIdx.x;
  if (idx >= kNQ * kNU) return;
  D12[idx] = D12t[idx] / lam[idx / kNU];
}

// aug (512x1024) = [E | I]
__global__ void k_init_aug(const float* __restrict__ E, float* __restrict__ aug)
{
  const int idx = blockIdx.x * blockDim.x + threadIdx.x;
  if (idx >= kNX * 2 * kNX) return;
  const int i = idx / (2 * kNX), j = idx % (2 * kNX);
  aug[idx] = (j < kNX) ? E[(size_t)i * kNX + j] : ((j - kNX) == i ? 1.0f : 0.0f);
}

// 512x512 Gauss-Jordan on global aug[512][1024]; 1 block, 1024 threads.
__global__ __launch_bounds__(1024) void k_gj512(float* __restrict__ aug)
{
  __shared__ float fcol[kNX];
  const int tid = threadIdx.x;
  for (int kp = 0; kp < kNX; ++kp) {
    const float pin = 1.0f / aug[(size_t)kp * 1024 + kp];
    if (tid < kNX) fcol[tid] = (tid == kp) ? 0.0f : aug[(size_t)tid * 1024 + kp];
    const float pr = aug[(size_t)kp * 1024 + tid] * pin;
    __syncthreads();
    aug[(size_t)kp * 1024 + tid] = pr;
    for (int i = 0; i < kNX; ++i)
      aug[(size_t)i * 1024 + tid] -= fcol[i] * pr;
    __threadfence();
    __syncthreads();
  }
}

// dst[c][r] = src[r][c]
__global__ void k_transpose(const float* __restrict__ src, float* __restrict__ dst,
                            int rows, int cols, int lds, int ldd)
{
  const int idx = blockIdx.x * blockDim.x + threadIdx.x;
  if (idx >= rows * cols) return;
  const int r = idx / cols, c = idx % cols;
  dst[(size_t)c * ldd + r] = src[(size_t)r * lds + c];
}

// ---------------------------------------------------------------------------
// Sequential tanh forward-substitution scan (32 batch rows per block in LDS):
//   w[:,k] = tanh((base[:,k] + w[:, :k] @ D11[k,:k]) / lam[k])
// ---------------------------------------------------------------------------
__global__ __launch_bounds__(256) void k_wscan(const float* __restrict__ base,
                                               const float* __restrict__ D11,
                                               const float* __restrict__ lam,
                                               float* __restrict__ w)
{
  __shared__ float wt[32][kNQ + 1];
  __shared__ float drow[kNQ];
  const int tid = threadIdx.x;
  const int r = tid >> 3, sub = tid & 7;
  const size_t gr = (size_t)blockIdx.x * 32 + r;
  for (int k = 0; k < kNQ; ++k) {
    drow[tid] = D11[(size_t)k * kNQ + tid];
    __syncthreads();
    float s = 0.0f;
    for (int j = sub; j < k; j += 8) s += wt[r][j] * drow[j];
    s += __shfl_xor(s, 1, 32);
    s += __shfl_xor(s, 2, 32);
    s += __shfl_xor(s, 4, 32);
    if (sub == 0) {
      const float v = (base[gr * kNQ + k] + s) / lam[k];
      wt[r][k] = tanhf(v);
    }
    __syncthreads();
  }
  for (int j = sub; j < kNQ; j += 8) w[gr * kNQ + j] = wt[r][j];
}

// ---------------------------------------------------------------------------
extern "C" void kernel_launch(void* const* d_in, const int* in_sizes, int n_in,
                              void* d_out, int out_size, void* d_ws, size_t ws_size,
                              hipStream_t stream)
{
  (void)in_sizes; (void)n_in; (void)out_size; (void)ws_size;
  const float* x    = (const float*)d_in[0];
  const float* u    = (const float*)d_in[1];
  const float* D12t = (const float*)d_in[2];
  const float* X3   = (const float*)d_in[3];
  const float* Y3   = (const float*)d_in[4];
  const float* Y1   = (const float*)d_in[5];
  const float* B2   = (const float*)d_in[6];
  const float* C2   = (const float*)d_in[7];
  const float* D21  = (const float*)d_in[8];
  const float* Xp   = (const float*)d_in[9];
  const float* Q    = (const float*)d_in[10];
  const float* S    = (const float*)d_in[11];
  const float* R    = (const float*)d_in[12];
  float* dx_out = (float*)d_out;
  float* y_out  = dx_out + (size_t)kNB * kNX;

  float* wsf = (float*)d_ws;
  size_t off = 0;
  auto alloc = [&](size_t n) { float* p = wsf + off; off += (n + 63) & ~(size_t)63; return p; };

  float* IpM    = alloc(kSD * kSD);
  float* ImM    = alloc(kSD * kSD);
  float* IpMinv = alloc(kSD * kSD);
  float* Mt     = alloc(kSD * kSD);
  float* Lq     = alloc(kSD * kSD);
  float* Qinv   = alloc(kSD * kSD);
  float* T1     = alloc(kSD * kSD);
  float* A2     = alloc(kSD * kSD);
  float* Lr     = alloc(kSD * kSD);
  float* T2     = alloc(kSD * kSD);
  float* D22w   = alloc(kSD * kSD);
  float* QD     = alloc(kSD * kSD);
  float* SD22   = alloc(kSD * kSD);
  float* Rc     = alloc(kSD * kSD);
  float* Rcinv  = alloc(kSD * kSD);
  float* G      = alloc(kSD * kSD);
  float* C2t    = alloc(kSD * kNX);
  float* D21t   = alloc(kSD * kNQ);
  float* V      = alloc((size_t)kN * kSD);
  float* vecC   = alloc((size_t)kN * kSD);
  float* Uu     = alloc((size_t)kN * kSD);
  float* Uc     = alloc((size_t)kN * kSD);
  float* H      = alloc((size_t)kN * kN);
  float* E      = alloc((size_t)kNX * kNX);
  float* lam    = alloc(kNQ);
  float* D11    = alloc((size_t)kNQ * kNQ);
  float* C1     = alloc((size_t)kNQ * kNX);
  float* D12    = alloc((size_t)kNQ * kNU);
  float* aug    = alloc((size_t)kNX * 2 * kNX);
  float* FT     = alloc((size_t)kNX * kNX);
  float* B1T    = alloc((size_t)kNQ * kNX);
  float* B2T    = alloc((size_t)kNU * kNX);
  float* EF     = alloc((size_t)kNX * kNX);
  float* EB1    = alloc((size_t)kNX * kNQ);
  float* EB2    = alloc((size_t)kNX * kNU);
  float* base   = alloc((size_t)kNB * kNQ);
  float* wbuf   = alloc((size_t)kNB * kNQ);

  auto gemmNT = [&](const float* A, const float* B, float* C, int M, int N, int K,
                    int lda, int ldb, int ldc, float alpha, float beta, float deps) {
    wmma_gemm_nt<<<dim3(N / TN, M / TM), dim3(256), 0, stream>>>(
        A, B, C, M, N, K, lda, ldb, ldc, alpha, beta, deps);
  };
  auto gemmS = [&](const float* A, const float* B, const float* Cin, float* C,
                   int M, int N, int K, int lda, int ldb, int ldci, int ldc,
                   float alpha, float beta, int tA, int tB, int tC) {
    gemm_small<<<dim3((M * N + 255) / 256), dim3(256), 0, stream>>>(
        A, B, Cin, C, M, N, K, lda, ldb, ldci, ldc, alpha, beta, tA, tB, tC);
  };
#define EW(kern, n, ...) kern<<<dim3(((n) + 255) / 256), dim3(256), 0, stream>>>(__VA_ARGS__)

  // ---- frame: small 64x64 algebra -----------------------------------------
  EW(k_build_M, kSD * kSD, X3, Y3, IpM, ImM);
  k_gj64<<<1, 256, 0, stream>>>(IpM, kSD, IpMinv);
  gemmS(ImM, IpMinv, nullptr, Mt, 64, 64, 64, 64, 64, 0, 64, 1, 0, 0, 0, 0);   // M_tilde
  k_chol64<<<1, 64, 0, stream>>>(Q, Lq);
  k_gj64<<<1, 256, 0, stream>>>(Q, kSD, Qinv);
  gemmS(Qinv, S, nullptr, T1, 64, 64, 64, 64, 64, 0, 64, 1, 0, 0, 1, 0);      // T1 = Q^-1 S^T
  gemmS(S, T1, R, A2, 64, 64, 64, 64, 64, 64, 64, -1, 1, 0, 0, 0);            // A2 = R - S T1
  k_chol64<<<1, 64, 0, stream>>>(A2, Lr);
  k_trsm64<<<1, 64, 0, stream>>>(Lq, Mt, T2);                                  // T2 = Lq^-1 Mt
  gemmS(T2, Lr, T1, D22w, 64, 64, 64, 64, 64, 64, 64, 1, 1, 0, 0, 0);         // D22
  gemmS(Q, D22w, nullptr, QD, 64, 64, 64, 64, 64, 0, 64, 1, 0, 0, 0, 0);
  gemmS(S, D22w, nullptr, SD22, 64, 64, 64, 64, 64, 0, 64, 1, 0, 0, 0, 0);
  gemmS(D22w, QD, R, Rc, 64, 64, 64, 64, 64, 64, 64, 1, 1, 1, 0, 0);          // R + D22^T Q D22
  EW(k_add_sym, kSD * kSD, Rc, SD22);                                          // + S D22 + (S D22)^T
  k_gj64<<<1, 256, 0, stream>>>(Rc, kSD, Rcinv);
  gemmS(D22w, Q, nullptr, G, 64, 64, 64, 64, 64, 0, 64, 1, 0, 1, 0, 0);       // G = D22^T Q
  gemmS(G, C2, nullptr, C2t, 64, 512, 64, 64, 512, 0, 512, 1, 0, 0, 0, 0);
  gemmS(S, C2, C2t, C2t, 64, 512, 64, 64, 512, 512, 512, 1, 1, 0, 0, 0);      // (G+S) C2
  gemmS(G, D21, D12t, D21t, 64, 256, 64, 64, 256, 64, 256, 1, -1, 0, 0, 1);   // G D21 - D12t^T
  EW(k_pack_VC, kN * kSD, C2t, D21t, B2, C2, D21, V, vecC);

  // ---- H = Xp Xp^T + eps I + V Rcap^-1 V^T - vecC Q vecC^T (WMMA) ---------
  gemmNT(V, Rcinv, Uu, kN, 64, 64, 64, 64, 64, 1, 0, 0);       // U  = V Rcinv (sym)
  gemmNT(vecC, Q, Uc, kN, 64, 64, 64, 64, 64, 1, 0, 0);        // Uc = vecC Q (sym)
  gemmNT(Xp, Xp, H, kN, kN, kN, kN, kN, kN, 1, 0, kEPS);
  gemmNT(Uu, V, H, kN, kN, 64, 64, 64, kN, 1, 1, 0);
  gemmNT(Uc, vecC, H, kN, kN, 64, 64, 64, kN, -1, 1, 0);

  // ---- slice H, build E/lam/D11/C1/D12 ------------------------------------
  EW(k_build_E, kNX * kNX, H, Y1, E);
  EW(k_lam, kNQ, H, lam);
  EW(k_build_D11, kNQ * kNQ, H, lam, D11);
  EW(k_build_C1, kNQ * kNX, H, C1);
  EW(k_build_D12, kNQ * kNU, D12t, lam, D12);

  // ---- Einv via Gauss-Jordan; EF/EB1/EB2 (WMMA with explicit transposes) --
  EW(k_init_aug, kNX * 2 * kNX, E, aug);
  k_gj512<<<1, 1024, 0, stream>>>(aug);
  const float* Einv = aug + kNX;  // right half, lda = 1024
  EW(k_transpose, kNX * kNX, H + (size_t)768 * kN, FT, 512, 512, kN, 512);
  EW(k_transpose, kNX * kNQ, H + (size_t)768 * kN + 512, B1T, 512, 256, kN, 512);
  EW(k_transpose, kNX * kNU, B2, B2T, 512, 64, 64, 512);
  gemmNT(Einv, FT, EF, 512, 512, 512, 1024, 512, 512, 1, 0, 0);
  gemmNT(Einv, B1T, EB1, 512, 256, 512, 1024, 512, 256, 1, 0, 0);
  gemmNT(Einv, B2T, EB2, 512, 64, 512, 1024, 512, 64, 1, 0, 0);

  // ---- batch phase (WMMA GEMMs + tanh scan) -------------------------------
  gemmNT(x, C1, base, kNB, kNQ, kNX, kNX, kNX, kNQ, 1, 0, 0);
  gemmNT(u, D12, base, kNB, kNQ, kNU, kNU, kNU, kNQ, 1, 1, 0);
  k_wscan<<<dim3(kNB / 32), dim3(256), 0, stream>>>(base, D11, lam, wbuf);

  gemmNT(x, EF, dx_out, kNB, kNX, kNX, kNX, kNX, kNX, 1, 0, 0);
  gemmNT(wbuf, EB1, dx_out, kNB, kNX, kNQ, kNQ, kNQ, kNX, 1, 1, 0);
  gemmNT(u, EB2, dx_out, kNB, kNX, kNU, kNU, kNU, kNX, 1, 1, 0);

  gemmNT(x, C2, y_out, kNB, kNY, kNX, kNX, kNX, kNY, 1, 0, 0);
  gemmNT(wbuf, D21, y_out, kNB, kNY, kNQ, kNQ, kNQ, kNY, 1, 1, 0);
  gemmNT(u, D22w, y_out, kNB, kNY, kNU, kNU, kNU, kNY, 1, 1, 0);
#undef EW
}